// SWM_FPEM_Loss_27882927685938
// MI455X (gfx1250) — compile-verified
//
#include <hip/hip_runtime.h>
#include <hip/hip_bf16.h>

// ---------------------------------------------------------------------------
// SWM_FPEM loss for MI455X (gfx1250, wave32).
// Memory-bound streaming + exact per-batch top-k via 4-level radix select on
// float bit patterns. CDNA5 paths: async global->LDS double-buffered staging
// (ASYNCcnt) and v_wmma_f32_16x16x4_f32 ones-matrix block reductions.
// ---------------------------------------------------------------------------

#define ALPHA_C 1.0f
#define NEG_POS_RATIO 3
#define EPB 8192   // elements per block per pass: 256 thr * 4 elem * 8 iters
#define ITERS 8

typedef __attribute__((ext_vector_type(2))) float v2f;
typedef __attribute__((ext_vector_type(8))) float v8f;
typedef int v4i_g __attribute__((vector_size(16)));  // matches builtin param

#if defined(__gfx1250__) && __has_builtin(__builtin_amdgcn_global_load_async_to_lds_b128)
#define USE_ASYNC 1
#else
#define USE_ASYNC 0
#endif

#if USE_ASYNC
typedef __attribute__((address_space(1))) v4i_g* g_v4i_ptr;
typedef __attribute__((address_space(3))) v4i_g* l_v4i_ptr;
__device__ __forceinline__ void async_b128(const float* g, float* l) {
  // global (AS1) -> LDS (AS3) 16B async copy; generic LDS addr low 32 bits are
  // the LDS offset (flat->LDS mapping truncates), so integer-cast is safe.
  __builtin_amdgcn_global_load_async_to_lds_b128(
      (g_v4i_ptr)(unsigned long long)g,
      (l_v4i_ptr)(unsigned int)(unsigned long long)l,
      0, 0);
}
__device__ __forceinline__ void wait_async0() {
#if __has_builtin(__builtin_amdgcn_s_wait_asynccnt)
  __builtin_amdgcn_s_wait_asynccnt(0);
#else
  asm volatile("s_wait_asynccnt 0" ::: "memory");
#endif
}
#endif

// Block-wide sum of 256 floats. WMMA path: 4x v_wmma_f32_16x16x4_f32 with
// all-ones A; sum over row 0 of D equals sum of all 64 B-register values per
// chunk (bijective layout => layout-agnostic). All waves execute redundantly
// so EXEC is all-ones everywhere (WMMA restriction).
__device__ __forceinline__ float block_reduce(float v, float* red) {
  const int tid = threadIdx.x;
  red[tid] = v;
  __syncthreads();
  float s;
#if __has_builtin(__builtin_amdgcn_wmma_f32_16x16x4_f32)
  {
    const int lane = tid & 31;
    v2f a; a[0] = 1.0f; a[1] = 1.0f;
    v8f c = {};
#pragma unroll
    for (int ch = 0; ch < 4; ++ch) {
      v2f bb;
      bb[0] = red[ch * 64 + lane];
      bb[1] = red[ch * 64 + 32 + lane];
      c = __builtin_amdgcn_wmma_f32_16x16x4_f32(false, a, false, bb,
                                                (short)0, c, false, false);
    }
    s = c[0];  // D row 0 (and row 8) column sums across 16-lane halves
#pragma unroll
    for (int off = 8; off; off >>= 1) s += __shfl_xor(s, off, 32);
  }
#else
  {
    float t = v;
#pragma unroll
    for (int off = 16; off; off >>= 1) t += __shfl_xor(t, off, 32);
    if ((tid & 31) == 0) red[tid >> 5] = t;
    __syncthreads();
    s = 0.0f;
    const int nw = (int)(blockDim.x >> 5);
    for (int i = 0; i < nw; ++i) s += red[i];
  }
#endif
  __syncthreads();  // red reused by subsequent calls
  return s;
}

// Workspace word layout (B = batch count), all 32-bit words:
//  [0]                      mse_sum
//  [1      .. 1+ B)         pos_sum[b]
//  [1+ B   .. 1+2B)         neg_cnt[b]   (exact integer in float)
//  [1+2B   .. 1+3B)         neg_sum[b]
//  [1+3B   .. 1+4B)         sum_gt[b]
//  [1+4B   .. 1+5B)         k_rem[b]
//  [1+5B   .. 1+6B)         neg_final[b]
//  [1+6B   .. 1+7B)         done[b]      (uint 0/1)
//  [1+7B   .. 1+8B)         prefix[b]    (uint, known high bits)
//  [1+8B   .. 1+8B+256B)    hist_cnt[b][256]
//  [..     .. 1+8B+512B)    hist_sum[b][256]

__global__ void init_ws(float* ws, int words) {
  int i = blockIdx.x * blockDim.x + threadIdx.x;
  if (i < words) ws[i] = 0.0f;
}

__global__ void pass_kernel(const float* __restrict__ y,
                            const float* __restrict__ o,
                            const float* __restrict__ w,
                            float* wsf, int B, long long N,
                            int blocksPerBatch, int level) {
  __shared__ float sh_cnt[256], sh_sum[256], red[256];
#if USE_ASYNC
  __shared__ float sy[2][1024], so2[2][1024], sw2[2][1024];
#endif
  const int tid = threadIdx.x;
  const int b = blockIdx.x / blocksPerBatch;
  const long long blk = blockIdx.x % blocksPerBatch;
  unsigned* U = (unsigned*)wsf;
  const int OFF_POS = 1, OFF_NEGCNT = 1 + B, OFF_NEGSUM = 1 + 2 * B;
  const int OFF_DONE = 1 + 6 * B, OFF_PREFIX = 1 + 7 * B;
  const int OFF_HCNT = 1 + 8 * B, OFF_HSUM = 1 + 8 * B + 256 * B;

  unsigned pref = 0;
  const int shift = 24 - 8 * level;
  if (level > 0) {
    if (U[OFF_DONE + b] != 0) return;  // uniform per block
    pref = U[OFF_PREFIX + b];
  }
  sh_cnt[tid] = 0.0f;
  sh_sum[tid] = 0.0f;
  __syncthreads();

  float lmse = 0.f, lpos = 0.f, lcnt = 0.f, lnsum = 0.f;
  const long long batchBase = (long long)b * N;
  const long long blkOff = blk * (long long)EPB;
  const bool full = (blkOff + EPB) <= N;

  auto process = [&](float yv, float ov, float wv) {
    float d = ov - yv;
    float m = d * d;
    lmse += m;
    bool pos = wv > 0.0f;
    if (pos) lpos += wv * m;
    if (ov > 0.0f && !pos) {
      unsigned bits = __float_as_uint(m);  // m >= 0: bits order == value order
      if (level == 0) {
        lcnt += 1.0f;
        lnsum += m;
        unsigned bkt = bits >> 24;
        atomicAdd(&sh_cnt[bkt], 1.0f);
        atomicAdd(&sh_sum[bkt], m);
      } else if ((bits >> (shift + 8)) == (pref >> (shift + 8))) {
        unsigned bkt = (bits >> shift) & 255u;
        atomicAdd(&sh_cnt[bkt], 1.0f);
        atomicAdd(&sh_sum[bkt], m);
      }
    }
  };

#if USE_ASYNC
  if (full) {
    auto stage = [&](int it) {
      long long off = blkOff + (long long)it * 1024 + tid * 4;
      int s = it & 1;
      async_b128(y + batchBase + off, &sy[s][tid * 4]);
      async_b128(o + batchBase + off, &so2[s][tid * 4]);
      async_b128(w + batchBase + off, &sw2[s][tid * 4]);
    };
    stage(0);
    for (int it = 0; it < ITERS; ++it) {
      wait_async0();
      __syncthreads();
      if (it + 1 < ITERS) stage(it + 1);  // prefetch into other buffer
      int s = it & 1;
#pragma unroll
      for (int e = 0; e < 4; ++e)
        process(sy[s][tid * 4 + e], so2[s][tid * 4 + e], sw2[s][tid * 4 + e]);
      __syncthreads();
    }
  } else
#endif
  {
    for (int it = 0; it < ITERS; ++it) {
      long long off = blkOff + (long long)it * 1024 + tid * 4;
      if (off + 4 <= N) {
        float4 yv = *(const float4*)(y + batchBase + off);
        float4 ov = *(const float4*)(o + batchBase + off);
        float4 wv = *(const float4*)(w + batchBase + off);
        process(yv.x, ov.x, wv.x);
        process(yv.y, ov.y, wv.y);
        process(yv.z, ov.z, wv.z);
        process(yv.w, ov.w, wv.w);
      } else {
        for (int e = 0; e < 4; ++e) {
          long long i = off + e;
          if (i < N) process(y[batchBase + i], o[batchBase + i], w[batchBase + i]);
        }
      }
    }
  }
  __syncthreads();

  if (sh_cnt[tid] != 0.0f) {
    atomicAdd(&wsf[OFF_HCNT + b * 256 + tid], sh_cnt[tid]);
    atomicAdd(&wsf[OFF_HSUM + b * 256 + tid], sh_sum[tid]);
  }
  if (level == 0) {
    float tm = block_reduce(lmse, red);
    float tp = block_reduce(lpos, red);
    float tc = block_reduce(lcnt, red);
    float tn = block_reduce(lnsum, red);
    if (tid == 0) {
      atomicAdd(&wsf[0], tm);
      atomicAdd(&wsf[OFF_POS + b], tp);
      atomicAdd(&wsf[OFF_NEGCNT + b], tc);
      atomicAdd(&wsf[OFF_NEGSUM + b], tn);
    }
  }
}

__global__ void select_kernel(float* wsf, const float* total_size, int B,
                              long long N, int level) {
  unsigned* U = (unsigned*)wsf;
  const int b = blockIdx.x;
  const int OFF_NEGCNT = 1 + B, OFF_NEGSUM = 1 + 2 * B, OFF_SUMGT = 1 + 3 * B;
  const int OFF_KREM = 1 + 4 * B, OFF_FINAL = 1 + 5 * B;
  const int OFF_DONE = 1 + 6 * B, OFF_PREFIX = 1 + 7 * B;
  const int OFF_HCNT = 1 + 8 * B, OFF_HSUM = 1 + 8 * B + 256 * B;
  float* cnt = wsf + OFF_HCNT + b * 256;
  float* sm = wsf + OFF_HSUM + b * 256;

  if (threadIdx.x == 0 && U[OFF_DONE + b] == 0) {
    bool finished = false;
    float final_v = 0.0f;
    if (level == 0) {
      float ts = total_size[b];
      long long k = (long long)(int)ts * NEG_POS_RATIO;
      if (k > N) k = N;
      if (ts <= 0.0f) {
        finished = true;  // per_b forced to 0 in final kernel anyway
      } else if ((float)k >= wsf[OFF_NEGCNT + b]) {
        finished = true;
        final_v = wsf[OFF_NEGSUM + b];  // fewer negatives than k: take all
      } else {
        wsf[OFF_KREM + b] = (float)k;
        wsf[OFF_SUMGT + b] = 0.0f;
        U[OFF_PREFIX + b] = 0u;
      }
    }
    if (!finished) {
      float krem = wsf[OFF_KREM + b];
      float crun = 0.0f, srun = 0.0f;
      int j = 255;
      for (; j >= 0; --j) {  // descending: high buckets = largest values
        float c = cnt[j];
        if (crun + c < krem) { crun += c; srun += sm[j]; }
        else break;
      }
      const int shift = 24 - 8 * level;
      if (j < 0) {  // defensive: everything consumed
        finished = true;
        final_v = wsf[OFF_SUMGT + b] + srun;
      } else {
        float c = cnt[j];
        if (crun + c == krem) {  // bucket j fully included: exact k reached
          finished = true;
          final_v = wsf[OFF_SUMGT + b] + srun + sm[j];
        } else if (level == 3) {  // all 32 bits fixed: exact threshold value
          unsigned bits = U[OFF_PREFIX + b] | (unsigned)j;
          finished = true;
          final_v = wsf[OFF_SUMGT + b] + srun +
                    (krem - crun) * __uint_as_float(bits);
        } else {  // refine into bucket j at next byte
          U[OFF_PREFIX + b] |= ((unsigned)j) << shift;
          wsf[OFF_KREM + b] = krem - crun;
          wsf[OFF_SUMGT + b] += srun;
        }
      }
    }
    if (finished) {
      U[OFF_DONE + b] = 1u;
      wsf[OFF_FINAL + b] = final_v;
    }
  }
  __syncthreads();
  for (int t = threadIdx.x; t < 256; t += blockDim.x) {  // reset for next level
    cnt[t] = 0.0f;
    sm[t] = 0.0f;
  }
}

__global__ void final_kernel(const float* wsf, const float* total_size,
                             float* outp, int B, long long N) {
  const int OFF_POS = 1, OFF_FINAL = 1 + 5 * B;
  __shared__ float s[256];
  const int t = threadIdx.x;
  float per = 0.0f;
  if (t < B) {
    float ts = total_size[t];
    if (ts > 0.0f)
      per = (ALPHA_C * wsf[OFF_POS + t] + wsf[OFF_FINAL + t]) / ts;
  }
  s[t] = per;
  __syncthreads();
  if (t == 0) {
    float tot = 0.0f;
    for (int i = 0; i < B && i < 256; ++i) tot += s[i];
    float mean_mse = wsf[0] / (float)((double)B * (double)N);
    outp[0] = (tot / (float)B + mean_mse) * 10.0f;
  }
}

extern "C" void kernel_launch(void* const* d_in, const int* in_sizes, int n_in,
                              void* d_out, int out_size, void* d_ws, size_t ws_size,
                              hipStream_t stream) {
  const float* y = (const float*)d_in[0];
  const float* o = (const float*)d_in[1];
  const float* w = (const float*)d_in[2];
  const float* ts = (const float*)d_in[3];

  const int B = in_sizes[3];                 // total_size is [B,1]
  const long long total = (long long)in_sizes[0];
  const long long N = total / B;

  float* ws = (float*)d_ws;
  const int words = 1 + 8 * B + 512 * B;     // ~33 KB for B=16
  init_ws<<<(words + 255) / 256, 256, 0, stream>>>(ws, words);

  const int bpb = (int)((N + EPB - 1) / EPB);
  dim3 grid((unsigned)(B * bpb));
  for (int lvl = 0; lvl < 4; ++lvl) {
    pass_kernel<<<grid, 256, 0, stream>>>(y, o, w, ws, B, N, bpb, lvl);
    select_kernel<<<B, 256, 0, stream>>>(ws, ts, B, N, lvl);
  }
  final_kernel<<<1, 256, 0, stream>>>(ws, ts, (float*)d_out, B, N);
}